// Logalike_3599182594864
// MI455X (gfx1250) — compile-verified
//
#include <hip/hip_runtime.h>
#include <hip/hip_bf16.h>
#include <math.h>

#define NUM_CELLS  512
#define NUM_SITES  32
#define NUM_STATES 24
#define DIMV       16
#define KDIM       56            // NUM_SITES + NUM_STATES
#define KP         64            // padded matrix size for WMMA tiling
#define S2         80            // pair-row stride in float2 units (bank-tuned)

typedef __attribute__((ext_vector_type(2))) float v2f;
typedef __attribute__((ext_vector_type(8))) float v8f;

// Matrix storage: pair layout. Element M[r][c] lives at
//   ((float*)P)[ ((r>>1)*S2 + c)*2 + (r&1) ]
// i.e. P[(r>>1)*S2 + c] = float2( M[2m][c], M[2m+1][c] ).
// A WMMA B-fragment (rows ka, ka+1 at column col, ka even) is exactly
// P[(ka>>1)*S2 + col] -> one ds_load_b64 into an even-aligned pair.

// ---------------------------------------------------------------------------
// Phase 1: hyperbolic distances -> t_c = dist_c / 2
// ---------------------------------------------------------------------------
__global__ void dist_kernel(const float* __restrict__ X,
                            const float* __restrict__ rho_p,
                            const int*   __restrict__ i_p,
                            float*       __restrict__ t_out) {
    int c = blockIdx.x * blockDim.x + threadIdx.x;
    if (c >= NUM_CELLS) return;
    const int   i   = *i_p;
    const float rho = *rho_p;
    const float* xi = X + (size_t)i * (DIMV + 1);
    const float* xc = X + (size_t)c * (DIMV + 1);
    float inner = -xi[0] * xc[0];
    #pragma unroll
    for (int d = 1; d <= DIMV; ++d) inner += xc[d] * xi[d];
    float arg  = fmaxf(-inner / rho, 1.0f + 1e-6f);
    float dist = sqrtf(rho) * acoshf(arg);
    t_out[c] = 0.5f * dist;
}

// ---------------------------------------------------------------------------
// 64x64x64 GEMM: A as per-wave register fragments, B/D in pair-layout LDS.
//   D = scaleOut * (A @ B)   (+ I if addI)
// ---------------------------------------------------------------------------
__device__ __forceinline__ void mm_fragA(v2f* __restrict__ Dp,
                                         const v2f afr[16],
                                         const v2f* __restrict__ Bp,
                                         float scaleOut, bool addI,
                                         int wave, int half, int lw) {
    #pragma unroll
    for (int tn = 0; tn < 4; ++tn) {
        v8f acc = {0.f, 0.f, 0.f, 0.f, 0.f, 0.f, 0.f, 0.f};
        const int col = tn * 16 + lw;
        #pragma unroll
        for (int kb = 0; kb < 16; ++kb) {
            // B rows {4kb+2h, 4kb+2h+1} at column col == pair entry (2kb+h, col)
            v2f b = Bp[(2 * kb + half) * S2 + col];
            acc = __builtin_amdgcn_wmma_f32_16x16x4_f32(
                false, afr[kb], false, b, (short)0, acc, false, false);
        }
        #pragma unroll
        for (int r = 0; r < 8; r += 2) {
            const int row = wave * 16 + r + 8 * half;     // even
            float v0 = acc[r]     * scaleOut;
            float v1 = acc[r + 1] * scaleOut;
            if (addI) {
                if (row     == col) v0 += 1.0f;
                if (row + 1 == col) v1 += 1.0f;
            }
            v2f d; d.x = v0; d.y = v1;
            Dp[(row >> 1) * S2 + col] = d;                // one ds_store_b64
        }
    }
}

// ---------------------------------------------------------------------------
// Phase 2: one block (128 threads = 4 waves) per (cell c, site s).
//   E = expm(t_c * Q_s) via scaling & squaring + order-8 Taylor (Horner),
//   then cur[c,s] = (1/NUM_STATES) * sum_a W[a] * E[a,ci] * E[a,cj],
//   logcur[c*S+s] = (c==i) ? 0 : log(cur).
// ---------------------------------------------------------------------------
__global__ __launch_bounds__(128) void expm_site_kernel(
        const float* __restrict__ Q,
        const int*   __restrict__ cm,
        const int*   __restrict__ i_p,
        const float* __restrict__ t_arr,
        float*       __restrict__ logcur) {
    __shared__ v2f   P0[(KP / 2) * S2];
    __shared__ v2f   P1[(KP / 2) * S2];
    __shared__ float red[128];
    __shared__ float s_scl[2];          // [0] = invs = exp2(-nsq); [1] = nsq

    const int blk  = blockIdx.x;
    const int c    = blk >> 5;          // / NUM_SITES
    const int s    = blk & 31;          // % NUM_SITES
    const int tid  = threadIdx.x;
    const int lane = tid & 31;
    const int wave = tid >> 5;
    const int half = lane >> 4;
    const int lw   = lane & 15;
    const int arow = wave * 16 + lw;    // the A-matrix row this lane owns

    // ---- load A = t * Q_s fragments straight into registers (zero-padded) ----
    const float  t  = t_arr[c];
    const float* Qs = Q + (size_t)s * KDIM * KDIM;
    v2f afr[16];
    float rsum = 0.f;                   // this lane's partial |row| sum
    #pragma unroll
    for (int kb = 0; kb < 16; ++kb) {
        const int ka = 4 * kb + 2 * half;
        float ax = 0.f, ay = 0.f;
        if (arow < KDIM) {
            if (ka     < KDIM) ax = t * Qs[arow * KDIM + ka];
            if (ka + 1 < KDIM) ay = t * Qs[arow * KDIM + ka + 1];
        }
        afr[kb].x = ax;
        afr[kb].y = ay;
        rsum += fabsf(ax) + fabsf(ay);
    }

    // ---- inf-norm -> number of squarings (theta = 0.5) ----
    red[tid] = rsum;                    // lanes l and l+16 of a wave share a row
    __syncthreads();
    if (tid == 0) {
        float m = 0.f;
        for (int w = 0; w < 4; ++w)
            for (int l = 0; l < 16; ++l)
                m = fmaxf(m, red[w * 32 + l] + red[w * 32 + l + 16]);
        int nsq = 0;
        if (m > 0.5f) nsq = (int)ceilf(log2f(2.0f * m));   // ceil(log2(m/0.5))
        if (nsq > 24) nsq = 24;
        s_scl[0] = exp2f(-(float)nsq);
        s_scl[1] = (float)nsq;
    }
    __syncthreads();
    const float invs = s_scl[0];
    const int   nsq  = (int)s_scl[1];

    // ---- Horner init: T0 = I + (invs/8) * A, written from the frags ----
    {
        float* T0f = (float*)P0;
        const float sc = invs * 0.125f;
        const int   rb = ((arow >> 1) * S2) * 2 + (arow & 1);  // row base (floats)
        #pragma unroll
        for (int kb = 0; kb < 16; ++kb) {
            const int ka = 4 * kb + 2 * half;
            T0f[rb + 2 * ka]       = sc * afr[kb].x + ((arow == ka    ) ? 1.0f : 0.0f);
            T0f[rb + 2 * (ka + 1)] = sc * afr[kb].y + ((arow == ka + 1) ? 1.0f : 0.0f);
        }
    }
    __syncthreads();

    v2f* Tc = P0;
    v2f* Tn = P1;
    // ---- Horner: T <- I + (invs/k) * (A @ T), k = 7..1  (order-8 Taylor) ----
    for (int k = 7; k >= 1; --k) {
        mm_fragA(Tn, afr, Tc, invs / (float)k, true, wave, half, lw);
        __syncthreads();
        v2f* tmp = Tc; Tc = Tn; Tn = tmp;
    }
    // ---- squarings: E <- E @ E (re-load A frags from Tc each round) ----
    for (int q = 0; q < nsq; ++q) {
        const float* Tf = (const float*)Tc;
        const int    rb = ((arow >> 1) * S2) * 2 + (arow & 1);
        v2f efr[16];
        #pragma unroll
        for (int kb = 0; kb < 16; ++kb) {
            const int ka = 4 * kb + 2 * half;
            efr[kb].x = Tf[rb + 2 * ka];
            efr[kb].y = Tf[rb + 2 * (ka + 1)];
        }
        mm_fragA(Tn, efr, Tc, 1.0f, false, wave, half, lw);
        __syncthreads();
        v2f* tmp = Tc; Tc = Tn; Tn = tmp;
    }

    // ---- feasible-ancestor weighted column product ----
    const int i0 = *i_p;
    const int si = cm[i0 * NUM_SITES + s];
    const int sj = cm[c  * NUM_SITES + s];
    const int ci = (si == 0) ? s : ((si == -1) ? (KDIM - 1) : si + NUM_SITES - 1);
    const int cj = (sj == 0) ? s : ((sj == -1) ? (KDIM - 1) : sj + NUM_SITES - 1);

    float val = 0.f;
    if (tid < KDIM) {
        const int a = tid;
        float w = (a == s) ? 1.0f : 0.0f;               // base: ancestor = site
        const bool si_m = (si == -1), sj_m = (sj == -1);
        if (si_m && sj_m) {                             // both missing: tail states
            if (a >= NUM_SITES) w += 1.0f;
        } else if (si_m != sj_m) {                      // one missing, other > 0
            const int other = si_m ? sj : si;
            if (other > 0) {
                const int aon  = other - 1;
                const int aidx = (aon == 0) ? s : aon + NUM_SITES - 1;
                if (a == aidx) w += 1.0f;
            }
        } else if (si == sj && si > 0) {                // equal nonzero states
            if (a == ci) w += 1.0f;
        }
        const float* Ef = (const float*)Tc;
        const int    ab = ((a >> 1) * S2) * 2 + (a & 1);
        val = w * Ef[ab + 2 * ci] * Ef[ab + 2 * cj];
    }
    __syncthreads();                    // red[] reused; protect RAW
    red[tid] = (tid < KDIM) ? val : 0.f;
    __syncthreads();
    if (tid == 0) {
        float sum = 0.f;
        for (int j = 0; j < KDIM; ++j) sum += red[j];
        const float cur = sum * (1.0f / (float)NUM_STATES);
        logcur[blk] = (c == i0) ? 0.0f : logf(cur);
    }
}

// ---------------------------------------------------------------------------
// Phase 3: deterministic reduction of the 16384 log terms -> scalar
// ---------------------------------------------------------------------------
__global__ void reduce_kernel(const float* __restrict__ logcur,
                              float* __restrict__ out) {
    __shared__ float sm[256];
    float acc = 0.f;
    for (int idx = threadIdx.x; idx < NUM_CELLS * NUM_SITES; idx += 256)
        acc += logcur[idx];
    sm[threadIdx.x] = acc;
    __syncthreads();
    for (int st = 128; st > 0; st >>= 1) {
        if (threadIdx.x < st) sm[threadIdx.x] += sm[threadIdx.x + st];
        __syncthreads();
    }
    if (threadIdx.x == 0) out[0] = sm[0];
}

// ---------------------------------------------------------------------------
extern "C" void kernel_launch(void* const* d_in, const int* in_sizes, int n_in,
                              void* d_out, int out_size, void* d_ws, size_t ws_size,
                              hipStream_t stream) {
    const float* X   = (const float*)d_in[0];   // [512, 17]
    const float* Q   = (const float*)d_in[1];   // [32, 56, 56]
    const float* rho = (const float*)d_in[2];   // scalar
    const int*   cm  = (const int*)  d_in[3];   // [512, 32]
    const int*   ip  = (const int*)  d_in[4];   // scalar
    float* out = (float*)d_out;

    float* t_ws   = (float*)d_ws;               // 512 floats
    float* logcur = t_ws + NUM_CELLS;           // 16384 floats

    dist_kernel<<<(NUM_CELLS + 127) / 128, 128, 0, stream>>>(X, rho, ip, t_ws);
    expm_site_kernel<<<NUM_CELLS * NUM_SITES, 128, 0, stream>>>(Q, cm, ip, t_ws, logcur);
    reduce_kernel<<<1, 256, 0, stream>>>(logcur, out);
}